// GAT_31653908971645
// MI455X (gfx1250) — compile-verified
//
#include <hip/hip_runtime.h>
#include <hip/hip_bf16.h>
#include <cstdint>

typedef __attribute__((ext_vector_type(2))) float v2f;
typedef __attribute__((ext_vector_type(8))) float v8f;

#define N_NODES  30000
#define E_REAL   480000
#define E_TOT    (E_REAL + N_NODES)      // edges + self loops
#define NEG_SLOPE 0.2f
#define NPAD     30016                   // N rounded up for WMMA tiles

// ---------- helpers ---------------------------------------------------------
__device__ __forceinline__ float lrelu(float v) { return v > 0.f ? v : NEG_SLOPE * v; }
// monotonic float<->int key for atomicMax on floats
__device__ __forceinline__ int   f2key(float f) { int x = __float_as_int(f); return x >= 0 ? x : (x ^ 0x7FFFFFFF); }
__device__ __forceinline__ float key2f(int k)   { return __int_as_float(k >= 0 ? k : (k ^ 0x7FFFFFFF)); }

// ---------- GEMM: Hout[N,HC] = X[N,K] * W[K,HC], f32 WMMA 16x16x4 -----------
// Block = 128 threads = 4 waves. Block tile = 64 rows x 64 cols.
// Wave w owns rows [w*16, w*16+16) and ALL 64 cols (4 WMMA tiles), so one A
// fragment feeds 4 WMMAs. The 64-col slab of W is staged in LDS per 32-K tile,
// pair-interleaved so a B fragment ({W[k][n], W[k+1][n]}) is one ds_load_b64.
// A-fragment layout (ISA 7.12.2, 32-bit A 16x4): lane m=lane&15 -> row m;
// lanes<16 carry K={0,1}, lanes>=16 carry K={2,3}.
__global__ void __launch_bounds__(128)
gemm_wmma_f32(const float* __restrict__ X, const float* __restrict__ W,
              float* __restrict__ Hout, int Nrows, int K, int Ncols) {
  __shared__ float Bs[16 * 128];                 // 32-K x 64-col tile, 8 KB
  const int tid  = threadIdx.x;
  const int wave = tid >> 5;
  const int lane = tid & 31;
  const int rowT = blockIdx.x * 64 + wave * 16;  // wave's 16-row tile
  const int colB = blockIdx.y * 64;              // block's 64-col slab
  const int m    = lane & 15;
  const int koff = (lane >> 4) << 1;             // 0 | 2
  const int row  = rowT + m;
  const int rowOK = row < Nrows;
  const float msk = rowOK ? 1.0f : 0.0f;         // branchless tail mask (EXEC full)
  const float* xrow = X + (size_t)(rowOK ? row : 0) * K;

  // cooperative W-load mapping: thread t -> W row (k0+kk), 16 cols from nb
  const int kk = tid >> 2;                       // 0..31
  const int nb = (tid & 3) << 4;                 // 0,16,32,48
  float* bs = Bs + ((kk >> 1) * 128 + (kk & 1)); // pair-interleaved base

  v8f acc0 = {}, acc1 = {}, acc2 = {}, acc3 = {};
  for (int k0 = 0; k0 < K; k0 += 32) {
    // ---- stage B tile: Bs[k/2][n*2 + (k&1)] = W[k0+k][colB+n] ----
    const float* wr = W + (size_t)(k0 + kk) * Ncols + colB + nb;
    #pragma unroll
    for (int j = 0; j < 16; j += 4) {
      float4 w4 = *(const float4*)(wr + j);
      bs[(nb + j + 0) * 2] = w4.x;
      bs[(nb + j + 1) * 2] = w4.y;
      bs[(nb + j + 2) * 2] = w4.z;
      bs[(nb + j + 3) * 2] = w4.w;
    }
    __syncthreads();
    // ---- 8 WMMA K-steps over the tile ----
    #pragma unroll
    for (int kq = 0; kq < 32; kq += 4) {
      v2f a = *(const v2f*)(xrow + k0 + kq + koff);   // one b64 load
      a.x *= msk; a.y *= msk;
      const float* bp = Bs + ((kq + koff) >> 1) * 128 + (m << 1);
      v2f b0 = *(const v2f*)(bp + 0);                 // ds_load_b64 each
      v2f b1 = *(const v2f*)(bp + 32);
      v2f b2 = *(const v2f*)(bp + 64);
      v2f b3 = *(const v2f*)(bp + 96);
      acc0 = __builtin_amdgcn_wmma_f32_16x16x4_f32(false, a, false, b0, (short)0, acc0, false, false);
      acc1 = __builtin_amdgcn_wmma_f32_16x16x4_f32(false, a, false, b1, (short)0, acc1, false, false);
      acc2 = __builtin_amdgcn_wmma_f32_16x16x4_f32(false, a, false, b2, (short)0, acc2, false, false);
      acc3 = __builtin_amdgcn_wmma_f32_16x16x4_f32(false, a, false, b3, (short)0, acc3, false, false);
    }
    __syncthreads();
  }
  // ---- store: C/D layout: VGPR i -> row i (lanes<16) / i+8 (lanes>=16) ----
  const int rbase = rowT + ((lane >> 4) << 3);
  #pragma unroll
  for (int i = 0; i < 8; ++i) {
    int r = rbase + i;
    if (r < Nrows) {
      float* hr = Hout + (size_t)r * Ncols + colB + m;
      hr[0]  = acc0[i];
      hr[16] = acc1[i];
      hr[32] = acc2[i];
      hr[48] = acc3[i];
    }
  }
}

// ---------- alpha_src / alpha_dst: [N,H] dots over C ------------------------
__global__ void node_alpha(const float* __restrict__ Hf,
                           const float* __restrict__ a_s,
                           const float* __restrict__ a_d,
                           float* __restrict__ out_s, float* __restrict__ out_d,
                           int N, int H, int C) {
  int t = blockIdx.x * blockDim.x + threadIdx.x;
  if (t >= N * H) return;
  int nn = t / H, hh = t - nn * H;
  const float* hp = Hf + (size_t)nn * H * C + (size_t)hh * C;
  const float* as = a_s + hh * C;
  const float* ad = a_d + hh * C;
  float s0 = 0.f, s1 = 0.f;
  for (int c = 0; c < C; ++c) { float v = hp[c]; s0 += v * as[c]; s1 += v * ad[c]; }
  out_s[t] = s0; out_d[t] = s1;
}

__global__ void init_neg_inf_keys(int* __restrict__ p, int n) {
  int t = blockIdx.x * blockDim.x + threadIdx.x;
  if (t < n) p[t] = 0x807FFFFF;   // f2key(-inf)
}

// ---------- edge pass 1: segment max of leaky_relu logits -------------------
__global__ void edge_max(const int* __restrict__ ei,
                         const float* __restrict__ as, const float* __restrict__ ad,
                         int* __restrict__ emax, int H) {
  int e = blockIdx.x * blockDim.x + threadIdx.x;
  if (e >= E_TOT) return;
  int s, d;
  if (e < E_REAL) { s = ei[e]; d = ei[E_REAL + e]; } else { s = d = e - E_REAL; }
  for (int h = 0; h < H; ++h) {
    float v = lrelu(as[s * H + h] + ad[d * H + h]);
    atomicMax(&emax[d * H + h], f2key(v));
  }
}

// ---------- edge pass 2: exp(e - max) + segment-sum denominator -------------
__global__ void edge_exp(const int* __restrict__ ei,
                         const float* __restrict__ as, const float* __restrict__ ad,
                         const int* __restrict__ emax,
                         float* __restrict__ denom, float* __restrict__ eexp, int H) {
  int e = blockIdx.x * blockDim.x + threadIdx.x;
  if (e >= E_TOT) return;
  int s, d;
  if (e < E_REAL) { s = ei[e]; d = ei[E_REAL + e]; } else { s = d = e - E_REAL; }
  for (int h = 0; h < H; ++h) {
    float v  = lrelu(as[s * H + h] + ad[d * H + h]);
    float mx = key2f(emax[d * H + h]);
    float ex = expf(v - mx);
    eexp[(size_t)e * H + h] = ex;
    atomicAdd(&denom[d * H + h], ex);
  }
}

// ---------- edge pass 3: out[dst] += alpha * h[src], float4 per thread ------
__global__ void edge_aggregate(const int* __restrict__ ei,
                               const float* __restrict__ eexp,
                               const float* __restrict__ denom,
                               const float* __restrict__ Hf,
                               float* __restrict__ agg,
                               int H, int HC, int qshift, int cshift) {
  int t = blockIdx.x * blockDim.x + threadIdx.x;
  int e = t >> qshift;                 // edge id
  int q = t & ((1 << qshift) - 1);     // float4 slot within HC
  if (e >= E_TOT) return;
  int s, d;
  if (e < E_REAL) { s = ei[e]; d = ei[E_REAL + e]; } else { s = d = e - E_REAL; }
  int c0 = q << 2;
  int hh = c0 >> cshift;               // which head this chunk belongs to
  float w = eexp[(size_t)e * H + hh] / denom[d * H + hh];
  const float4 hv = *(const float4*)(Hf + (size_t)s * HC + c0);
  float* ap = agg + (size_t)d * HC + c0;
  atomicAdd(ap + 0, w * hv.x);
  atomicAdd(ap + 1, w * hv.y);
  atomicAdd(ap + 2, w * hv.z);
  atomicAdd(ap + 3, w * hv.w);
}

// ---------- bias + relu (layer4: H=1 so mean over heads == identity) --------
__global__ void bias_relu(const float* __restrict__ x, const float* __restrict__ b,
                          float* __restrict__ y, int total, int hcmask) {
  int t = blockIdx.x * blockDim.x + threadIdx.x;
  if (t >= total) return;
  float v = x[t] + b[t & hcmask];
  y[t] = v > 0.f ? v : 0.f;
}

// ---------- host-side layer driver ------------------------------------------
static void run_layer(const float* X, int K, const float* W,
                      const float* as_v, const float* ad_v, const float* bias,
                      int H, int C,
                      float* hbuf, float* aggbuf, float* outbuf,
                      float* alpha_s, float* alpha_d, int* emax, float* denom,
                      float* eexp, const int* ei, hipStream_t stream) {
  const int HC = H * C;
  // 1) GEMM (WMMA f32, LDS-staged W)
  dim3 ggrd((N_NODES + 63) / 64, HC / 64);
  gemm_wmma_f32<<<ggrd, dim3(128), 0, stream>>>(X, W, hbuf, N_NODES, K, HC);
  // 2) attention logits per node
  const int nh = N_NODES * H;
  node_alpha<<<(nh + 255) / 256, 256, 0, stream>>>(hbuf, as_v, ad_v,
                                                   alpha_s, alpha_d, N_NODES, H, C);
  // 3) re-init reduction buffers (deterministic every call)
  init_neg_inf_keys<<<(nh + 255) / 256, 256, 0, stream>>>(emax, nh);
  hipMemsetAsync(denom, 0, sizeof(float) * (size_t)nh, stream);
  hipMemsetAsync(aggbuf, 0, sizeof(float) * (size_t)N_NODES * HC, stream);
  // 4) scatter softmax
  edge_max<<<(E_TOT + 255) / 256, 256, 0, stream>>>(ei, alpha_s, alpha_d, emax, H);
  edge_exp<<<(E_TOT + 255) / 256, 256, 0, stream>>>(ei, alpha_s, alpha_d, emax,
                                                    denom, eexp, H);
  // 5) weighted message scatter-add
  const int qshift = __builtin_ctz(HC / 4);
  const int cshift = __builtin_ctz(C);
  long long tot = (long long)E_TOT << qshift;
  edge_aggregate<<<(unsigned)((tot + 255) / 256), 256, 0, stream>>>(
      ei, eexp, denom, hbuf, aggbuf, H, HC, qshift, cshift);
  // 6) bias + relu
  const int ne = N_NODES * HC;
  bias_relu<<<(ne + 255) / 256, 256, 0, stream>>>(aggbuf, bias, outbuf, ne, HC - 1);
}

extern "C" void kernel_launch(void* const* d_in, const int* in_sizes, int n_in,
                              void* d_out, int out_size, void* d_ws, size_t ws_size,
                              hipStream_t stream) {
  (void)in_sizes; (void)n_in; (void)out_size; (void)ws_size;
  const float* x   = (const float*)d_in[0];
  const int*   ei  = (const int*)d_in[1];     // [2, 480000] flattened, row0=src
  const float* W1  = (const float*)d_in[2];
  const float* a1s = (const float*)d_in[3];
  const float* a1d = (const float*)d_in[4];
  const float* b1  = (const float*)d_in[5];
  const float* W2  = (const float*)d_in[6];
  const float* a2s = (const float*)d_in[7];
  const float* a2d = (const float*)d_in[8];
  const float* b2  = (const float*)d_in[9];
  const float* W3  = (const float*)d_in[10];
  const float* a3s = (const float*)d_in[11];
  const float* a3d = (const float*)d_in[12];
  const float* b3  = (const float*)d_in[13];
  const float* W4  = (const float*)d_in[14];
  const float* a4s = (const float*)d_in[15];
  const float* a4d = (const float*)d_in[16];
  const float* b4  = (const float*)d_in[17];

  // workspace layout (~195 MB): three [NPAD,512] feature buffers + edge scratch
  float* ws = (float*)d_ws;
  const size_t NB = (size_t)NPAD * 512;
  float* bufA    = ws;
  float* bufB    = bufA + NB;          // GEMM output (reused each layer)
  float* bufC    = bufB + NB;
  float* alpha_s = bufC + NB;
  float* alpha_d = alpha_s + (size_t)N_NODES * 4;
  int*   emax    = (int*)(alpha_d + (size_t)N_NODES * 4);
  float* denom   = (float*)(emax + (size_t)N_NODES * 4);
  float* eexp    = denom + (size_t)N_NODES * 4;   // [E_TOT, 4]

  // L1: 256 -> 4x128 concat ; h=bufB, agg->bufA (finalize in place)
  run_layer(x,    256, W1, a1s, a1d, b1, 4, 128, bufB, bufA, bufA,
            alpha_s, alpha_d, emax, denom, eexp, ei, stream);
  // L2: 512 -> 4x128 concat
  run_layer(bufA, 512, W2, a2s, a2d, b2, 4, 128, bufB, bufC, bufC,
            alpha_s, alpha_d, emax, denom, eexp, ei, stream);
  // L3: 512 -> 4x64 concat
  run_layer(bufC, 512, W3, a3s, a3d, b3, 4, 64,  bufB, bufA, bufA,
            alpha_s, alpha_d, emax, denom, eexp, ei, stream);
  // L4: 256 -> 1x64, concat=False (H=1 => mean==identity), write d_out
  run_layer(bufA, 256, W4, a4s, a4d, b4, 1, 64,  bufB, bufC, (float*)d_out,
            alpha_s, alpha_d, emax, denom, eexp, ei, stream);
}